// ModelNew_4810363372028
// MI455X (gfx1250) — compile-verified
//
#include <hip/hip_runtime.h>

// C = triu( triu(A) @ triu(B) ), N = 4096, fp32, via V_WMMA_F32_16X16X4_F32.
// Inputs are stored already-triangular => dense block GEMM over live k-blocks
// is exact; only the output needs the upper-triangular mask.

typedef __attribute__((ext_vector_type(2))) float v2f;
typedef __attribute__((ext_vector_type(4))) float v4f;
typedef __attribute__((ext_vector_type(8))) float v8f;
typedef __attribute__((ext_vector_type(4))) int   v4i;

#define NMAT 4096
#define TM   128            // C tile rows per workgroup (= k-slab depth)
#define TN   256            // C tile cols per workgroup
#define LDA  132            // sA row stride (floats): lane bank stride 4 -> conflict-free
#define LDB  136            // sB row stride (floats): K+2 lanes land +16 banks -> conflict-free,
                            // and 136*4B fits ds_load_2addr_b32 offset range
#define SBH  (128 * LDB)    // one 128-col half of the B slab

#if __has_builtin(__builtin_amdgcn_global_load_async_to_lds_b128)
#define HAVE_ASYNC_LDS 1
#define GPTR(p) ((__attribute__((address_space(1))) v4i*)(p))
#define LPTR(p) ((__attribute__((address_space(3))) v4i*)(p))
#else
#define HAVE_ASYNC_LDS 0
#endif

__device__ __forceinline__ void wait_async_then_barrier() {
#if HAVE_ASYNC_LDS
#if __has_builtin(__builtin_amdgcn_s_wait_asynccnt)
    __builtin_amdgcn_s_wait_asynccnt(0);
#else
    asm volatile("s_wait_asynccnt 0" ::: "memory");
#endif
#endif
    __syncthreads();
}

__global__ __launch_bounds__(256)
void trimm_wmma_f32(const float* __restrict__ A,
                    const float* __restrict__ B,
                    float* __restrict__ C) {
    const int bjw = blockIdx.x;          // 0..15 : 256-wide col tile
    const int bi  = blockIdx.y;          // 0..31 : 128-tall row tile
    const int tid = threadIdx.x;

    const size_t gRow0 = (size_t)bi  * TM;
    const size_t gCol0 = (size_t)bjw * TN;

    // ---- Fully-dead tiles (every row > every col): write zeros. ----
    if (bi > 2 * bjw + 1) {
        v4f z = {0.f, 0.f, 0.f, 0.f};
#pragma unroll 4
        for (int i = 0; i < 32; ++i) {
            int idx = tid + (i << 8);            // 0..8191 over 128x64 float4s
            int row = idx >> 6;
            int c4  = (idx & 63) << 2;
            *(v4f*)(C + (gRow0 + row) * NMAT + gCol0 + c4) = z;
        }
        return;
    }

    // ---- LDS slabs (207 KB total: CDNA5 WGP has 320 KB) ----
    __shared__ float sA[TM * LDA];       // A[row][k]   : 67.6 KB
    __shared__ float sB[2 * SBH];        // B[half][k][col&127] : 139 KB

    const int lane = tid & 31;
    const int w    = tid >> 5;           // wave 0..7
    const int wr   = w >> 2;             // 0..1 -> 64-row strip
    const int wc   = w & 3;              // 0..3 -> 64-col strip
    const int rBase = wr * 64;
    const int cBase = wc * 64;
    const int m     = lane & 15;
    const int koff  = (lane >> 4) << 1;  // lanes 16-31 hold K=2,3

    v8f acc[4][4];
#pragma unroll
    for (int r = 0; r < 4; ++r)
#pragma unroll
        for (int c = 0; c < 4; ++c)
            acc[r][c] = (v8f){0.f, 0.f, 0.f, 0.f, 0.f, 0.f, 0.f, 0.f};

    // Per-wave LDS base pointers (constant across the k loop).
    const float* aBase[4];
#pragma unroll
    for (int r = 0; r < 4; ++r)
        aBase[r] = sA + (rBase + r * 16 + m) * LDA;
    const float* bBase[4];
#pragma unroll
    for (int c = 0; c < 4; ++c) {
        int col = cBase + c * 16 + m;                 // 0..255
        bBase[c] = sB + (col >> 7) * SBH + (col & 127);
    }

    // Live k-blocks: A(bi,kb) nonzero for kb >= bi; B(kb, this col tile)
    // nonzero for kb*128 <= gCol0+255  =>  kb <= 2*bjw+1.
    const int kbEnd = (2 * bjw + 1 < 31) ? (2 * bjw + 1) : 31;

    for (int kb = bi; kb <= kbEnd; ++kb) {
        __syncthreads();                 // previous compute done reading LDS
        const size_t kOfs = (size_t)kb * 128;
        const bool   more = (kb < kbEnd);

        // Stage A slab: 128x128 (64 floats/thread). Async global->LDS keeps
        // the staged data out of VGPRs (ASYNCcnt path on CDNA5).
#pragma unroll 4
        for (int i = 0; i < 16; ++i) {
            int idx = tid + (i << 8);
            int row = idx >> 5;                      // 0..127
            int c4  = (idx & 31) << 2;               // 0..124
            const float* pa = A + (gRow0 + row) * NMAT + kOfs + c4;
            float*       la = sA + row * LDA + c4;
#if HAVE_ASYNC_LDS
            __builtin_amdgcn_global_load_async_to_lds_b128(GPTR(pa), LPTR(la), 0, 0);
#else
            *(v4f*)la = *(const v4f*)pa;
#endif
            if (more) __builtin_prefetch(pa + 128, 0, 3);
        }
        // Stage B slab: 128x256 (128 floats/thread), split into two halves.
#pragma unroll 4
        for (int i = 0; i < 32; ++i) {
            int idx = tid + (i << 8);
            int row = idx >> 6;                      // 0..127
            int c4  = (idx & 63) << 2;               // 0..252
            const float* pb = B + (kOfs + row) * NMAT + gCol0 + c4;
            float*       lb = sB + (c4 >> 7) * SBH + row * LDB + (c4 & 127);
#if HAVE_ASYNC_LDS
            __builtin_amdgcn_global_load_async_to_lds_b128(GPTR(pb), LPTR(lb), 0, 0);
#else
            *(v4f*)lb = *(const v4f*)pb;
#endif
            if (more) __builtin_prefetch(pb + (size_t)128 * NMAT, 0, 3);
        }
        wait_async_then_barrier();

        // 32 k-steps of 4. Unroll by 4: DS offsets stay immediate while the
        // operand-buffer window keeps total VGPRs under the 256 MSB boundary.
#pragma unroll 4
        for (int k4 = 0; k4 < 32; ++k4) {
            const int kk = (k4 << 2) + koff;
            v2f a[4], b[4];
#pragma unroll
            for (int r = 0; r < 4; ++r)
                a[r] = *(const v2f*)(aBase[r] + kk);
#pragma unroll
            for (int c = 0; c < 4; ++c) {
                b[c].x = bBase[c][(size_t)kk * LDB];
                b[c].y = bBase[c][(size_t)(kk + 1) * LDB];
            }
#pragma unroll
            for (int r = 0; r < 4; ++r)
#pragma unroll
                for (int c = 0; c < 4; ++c)
                    acc[r][c] = __builtin_amdgcn_wmma_f32_16x16x4_f32(
                        false, a[r], false, b[c],
                        (short)0, acc[r][c], false, false);
        }
    }

    // ---- Store with upper-triangular mask (row <= col). ----
    const int mb = (lane >> 4) << 3;     // lanes 16-31 hold M = 8..15
#pragma unroll
    for (int r = 0; r < 4; ++r) {
#pragma unroll
        for (int c = 0; c < 4; ++c) {
            const size_t col = gCol0 + cBase + c * 16 + m;
#pragma unroll
            for (int v = 0; v < 8; ++v) {
                const size_t row = gRow0 + rBase + r * 16 + mb + v;
                C[row * NMAT + col] = (row <= col) ? acc[r][c][v] : 0.0f;
            }
        }
    }
}

extern "C" void kernel_launch(void* const* d_in, const int* in_sizes, int n_in,
                              void* d_out, int out_size, void* d_ws, size_t ws_size,
                              hipStream_t stream) {
    (void)in_sizes; (void)n_in; (void)out_size; (void)d_ws; (void)ws_size;
    const float* A = (const float*)d_in[0];
    const float* B = (const float*)d_in[1];
    float* C = (float*)d_out;
    dim3 grid(NMAT / TN, NMAT / TM);     // 16 x 32 tiles
    trimm_wmma_f32<<<grid, 256, 0, stream>>>(A, B, C);
}